// SubsampledRelativeAttention_1443109011856
// MI455X (gfx1250) — compile-verified
//
#include <hip/hip_runtime.h>

typedef __attribute__((ext_vector_type(2))) float v2f;
typedef __attribute__((ext_vector_type(8))) float v8f;

constexpr int kH = 8;
constexpr int kS = 256;
constexpr int kT = 1024;
constexpr int kD = 64;
constexpr int kRatio = 4;
constexpr int kQRows = 64;    // 16 sb * 4 r
constexpr int kWRows = 256;   // full s range -> full 256-row gather window
constexpr int kStr = 68;      // padded LDS row stride in floats (bank-conflict-free)

// out[b,t,s] = q[b,t,:] . W[h, 255 + s - t/4, :],  h = b % 8
//   W[h,j] = e1[h*256 + j]        for j <= 255
//   W[h,j] = e2[h*256 + (j-255)]  for j >= 256
__global__ __launch_bounds__(256)
void sra_wmma_f32_kernel(const float* __restrict__ q,
                         const float* __restrict__ e1,
                         const float* __restrict__ e2,
                         float* __restrict__ out)
{
    __shared__ float q_lds[kQRows * kStr];   // 17,408 B
    __shared__ float w_lds[kWRows * kStr];   // 69,632 B  (total 87,040 B < 320 KB/WGP)

    const int a     = blockIdx.x;            // 0..255 (t / 4)
    const int h     = blockIdx.y;            // 0..7   (head)
    const int jbase = 255 - a;               // combined-table window start

    const int tid = threadIdx.x;

    // ---- stage q block: row m = sb*4 + r  (64 rows x 64 cols) ----
    {
        const int m       = tid >> 2;        // 0..63
        const int quarter = tid & 3;         // 16 floats each
        const int sb = m >> 2;
        const int r  = m & 3;
        const int b  = sb * kH + h;
        const int t  = a * kRatio + r;
        const float4* src = (const float4*)(q + ((size_t)b * kT + t) * kD + quarter * 16);
        float4* dst = (float4*)(q_lds + m * kStr + quarter * 16);
        dst[0] = src[0]; dst[1] = src[1]; dst[2] = src[2]; dst[3] = src[3];
    }
    // ---- stage W window: 256 rows x 64 cols, local row l -> table row j = jbase + l ----
    // coalesced: consecutive lanes read consecutive float4s within a row
#pragma unroll
    for (int p = 0; p < 16; ++p) {
        const int f4 = p * 256 + tid;        // float4 index into 256x64 window
        const int l  = f4 >> 4;              // row 0..255
        const int c4 = f4 & 15;              // float4 column 0..15
        const int j  = jbase + l;
        const float* srow = (j <= 255) ? (e1 + (size_t)(h * kS + j) * kD)
                                       : (e2 + (size_t)(h * kS + (j - 255)) * kD);
        ((float4*)(w_lds + l * kStr))[c4] = ((const float4*)srow)[c4];
    }
    __syncthreads();

    const int lane    = tid & 31;
    const int wave    = tid >> 5;            // 0..7
    const int row_blk = wave & 3;            // 16 of the 64 q rows
    const int colgrp  = wave >> 2;           // 8 of the 16 col tiles

    const int mrow = lane & 15;              // tile row (A) / col (B) index
    const int koff = (lane >> 4) << 1;       // K sub-offset: 0 (lanes 0-15) or 2 (16-31)

    // ---- preload A fragments for all 16 K-steps (reused across 8 col tiles) ----
    v2f afrag[16];
    const float* qrow = q_lds + (row_blk * 16 + mrow) * kStr + koff;
#pragma unroll
    for (int kk = 0; kk < 16; ++kk)
        afrag[kk] = *(const v2f*)(qrow + kk * 4);

#pragma unroll
    for (int cc = 0; cc < 8; ++cc) {
        const int ct = colgrp * 8 + cc;      // col tile in [0,16)
        const float* wcol = w_lds + (ct * 16 + mrow) * kStr + koff;
        v8f acc = {};
#pragma unroll
        for (int kk = 0; kk < 16; ++kk) {
            v2f bfrag = *(const v2f*)(wcol + kk * 4);
            acc = __builtin_amdgcn_wmma_f32_16x16x4_f32(
                false, afrag[kk], false, bfrag, (short)0, acc, false, false);
        }
        // ---- store 16x16 D tile (lanes 0-15: M=v; lanes 16-31: M=8+v) ----
        const int s     = ct * 16 + (lane & 15);
        const int mbase = (lane >> 4) << 3;
#pragma unroll
        for (int v = 0; v < 8; ++v) {
            const int gm = row_blk * 16 + mbase + v;   // 0..63
            const int sb = gm >> 2;
            const int r  = gm & 3;
            const int b  = sb * kH + h;
            const int t  = a * kRatio + r;
            out[((size_t)b * kT + t) * kS + s] = acc[v];
        }
    }
}

extern "C" void kernel_launch(void* const* d_in, const int* in_sizes, int n_in,
                              void* d_out, int out_size, void* d_ws, size_t ws_size,
                              hipStream_t stream) {
    const float* q  = (const float*)d_in[0];
    const float* e1 = (const float*)d_in[1];
    const float* e2 = (const float*)d_in[2];
    float* out = (float*)d_out;

    dim3 grid(kS, kH);      // (a = t/4, head) — q staged exactly once from HBM
    dim3 block(256);        // 8 waves of 32
    hipLaunchKernelGGL(sra_wmma_f32_kernel, grid, block, 0, stream, q, e1, e2, out);
}